// CQTsmall_11725260718517
// MI455X (gfx1250) — compile-verified
//
#include <hip/hip_runtime.h>

typedef __attribute__((ext_vector_type(2))) float v2f;
typedef __attribute__((ext_vector_type(8))) float v8f;

#define BATCH    16
#define BPO      36
#define NFILT    72
#define WIDTH    271
#define KPAD     272      // 271 taps padded to multiple of 4 (tap 271 = 0)
#define PADC     135
#define OCTAVES  8
#define L0       524288
#define T_OUT    2048
#define NCH      288      // 8*36 output channel rows per real/imag half
#define HT       64       // filtfilt-FIR half width
#define GLEN     (2*HT+1) // 129 taps
#define IMP_LEN  192      // truncated IIR impulse response length

#if defined(__gfx1250__) && __has_builtin(__builtin_amdgcn_global_load_async_to_lds_b32)
#define USE_ASYNC_LDS 1
typedef __attribute__((address_space(1))) int as1_int;
typedef __attribute__((address_space(3))) int as3_int;
#endif

// ---------------------------------------------------------------------------
// Stage A: build the zero-phase FIR equivalent of filtfilt(butterworth N=10).
// g[k] = sum_n h[n]*h[n+|k|]  (autocorrelation of the IIR impulse response)
// ---------------------------------------------------------------------------
__global__ void fir_taps_kernel(const float* __restrict__ num,
                                const float* __restrict__ den,
                                float* __restrict__ G) {
  __shared__ float h[IMP_LEN];
  int tid = threadIdx.x;
  if (tid == 0) {
    float bb[11], aa[11];
    float a0 = den[0];
    for (int i = 0; i <= 10; ++i) { bb[i] = num[i] / a0; aa[i] = den[i] / a0; }
    for (int n = 0; n < IMP_LEN; ++n) {
      float v = (n <= 10) ? bb[n] : 0.0f;
      for (int i = 1; i <= 10; ++i) {
        int p = n - i;
        if (p >= 0) v -= aa[i] * h[p];
      }
      h[n] = v;
    }
  }
  __syncthreads();
  for (int k = tid; k <= HT; k += blockDim.x) {
    float s = 0.0f;
    for (int n = 0; n + k < IMP_LEN; ++n) s += h[n] * h[n + k];
    G[HT + k] = s;
    G[HT - k] = s;
  }
}

// ---------------------------------------------------------------------------
// Stage B: zero-phase FIR + decimate-by-2 through an LDS tile.
// y[b,m] = sum_{j=0..128} G[j] * x[b, 2m + j - 64]   (zeros outside [0,Lin))
// ---------------------------------------------------------------------------
__global__ void __launch_bounds__(256)
downsample_kernel(const float* __restrict__ xin, float* __restrict__ xout,
                  const float* __restrict__ G, int Lin, int Lout) {
  __shared__ float tile[2 * 256 + 2 * HT + 2];
  __shared__ float g[GLEN];
  int tid = threadIdx.x;
  int b   = blockIdx.y;
  int m0  = blockIdx.x * 256;
  const float* xb = xin + (size_t)b * Lin;
  int s0   = 2 * m0 - HT;
  int need = 2 * 256 + 2 * HT + 1;
  for (int i = tid; i < need; i += 256) {
    int s = s0 + i;
    tile[i] = (s >= 0 && s < Lin) ? xb[s] : 0.0f;
  }
  for (int i = tid; i < GLEN; i += 256) g[i] = G[i];
  __syncthreads();
  int m = m0 + tid;
  if (m < Lout) {
    float acc = 0.0f;
    int base = 2 * tid;
    for (int j = 0; j < GLEN; ++j) acc += g[j] * tile[base + j];
    xout[(size_t)b * Lout + m] = acc;
  }
}

// ---------------------------------------------------------------------------
// Stage C: CQT conv as WMMA f32 16x16x4 GEMM. Two wave32s per workgroup
// share one 16-filter kernel tile; each wave owns a 16-frame N-tile.
// D[m,n] = sum_k Kern[mtile*16+m, k] * sig[b, (t0+n)*hop + k - PADC]
// Signal window staged via GLOBAL_LOAD_ASYNC_TO_LDS (ASYNCcnt) when available.
// ---------------------------------------------------------------------------
__global__ void __launch_bounds__(64)
cqt_wmma_kernel(const float* __restrict__ sig, const float* __restrict__ kern,
                float* __restrict__ out, int Lsig, int hop, int blk) {
  __shared__ float lk[16 * KPAD];                 // 16 filter rows x 272 taps
  __shared__ float lx[31 * 256 + KPAD + 16];      // 32-frame window (max hop=256)
  int tid   = threadIdx.x;                        // 0..63
  int lane  = tid & 31;
  int w     = tid >> 5;                           // wave id: N-subtile
  int mtile = blockIdx.y, b = blockIdx.z;

  // --- Stage kernel tile (zero-pad rows >= 72 and tap 271) ---
  for (int i = tid; i < 16 * KPAD; i += 64) {
    int m = i / KPAD, k = i - m * KPAD;
    int row = mtile * 16 + m;
    lk[i] = (row < NFILT && k < WIDTH) ? kern[(size_t)row * WIDTH + k] : 0.0f;
  }

  // --- Stage 32-frame signal window ---
  const float* xb = sig + (size_t)b * Lsig;
  int tbase = blockIdx.x * 32;                    // first frame of the block
  int s0    = tbase * hop - PADC;
  int wlen  = 31 * hop + KPAD;
  bool interior = (s0 >= 0) && (s0 + wlen <= Lsig);
#if USE_ASYNC_LDS
  if (!interior) {                                // pre-zero only edge blocks
    for (int i = tid; i < wlen; i += 64) lx[i] = 0.0f;
    __syncthreads();
  }
  for (int i = tid; i < wlen; i += 64) {
    int s = s0 + i;
    if (s >= 0 && s < Lsig) {                     // EXEC-masked async DMA to LDS
      __builtin_amdgcn_global_load_async_to_lds_b32(
          (as1_int*)(xb + s), (as3_int*)(lx + i), 0, 0);
    }
  }
#if __has_builtin(__builtin_amdgcn_s_wait_asynccnt)
  __builtin_amdgcn_s_wait_asynccnt(0);
#else
  asm volatile("s_wait_asynccnt 0x0" ::: "memory");
#endif
#else
  for (int i = tid; i < wlen; i += 64) {
    int s = s0 + i;
    lx[i] = (s >= 0 && s < Lsig) ? xb[s] : 0.0f;
  }
#endif
  __syncthreads();

  // ISA 7.12.2 layouts: A 16x4 f32 -> lane<16 holds K=k0,k0+1; lane>=16 K=k0+2,k0+3; M=lane%16.
  int sel2 = (lane >> 4) << 1;   // 0 or 2
  int mrow = lane & 15;          // A's M row / B's N column
  int nloc = w * 16 + mrow;      // frame within the 32-frame window
  v8f c = {};
  for (int kt = 0; kt < KPAD / 4; ++kt) {
    int k0 = kt * 4 + sel2;
    v2f a, bf;
    a.x = lk[mrow * KPAD + k0];
    a.y = lk[mrow * KPAD + k0 + 1];
    int off = nloc * hop + k0;
    bf.x = lx[off];
    bf.y = lx[off + 1];
    // 8 args: (neg_a, A, neg_b, B, c_mod, C, reuse_a, reuse_b)
    c = __builtin_amdgcn_wmma_f32_16x16x4_f32(
        false, a, false, bf, (short)0, c, false, false);
  }

  // C/D layout: VGPR r holds row (lane/16)*8 + r, column N = lane%16.
  // addr(row) = base0 + row*T_OUT + (row>=36 ? (NCH-BPO)*T_OUT : 0)  -- no div/mod
  int t      = tbase + w * 16 + (lane & 15);
  int mbase  = mtile * 16 + (lane >> 4) * 8;
  size_t base0 = (((size_t)b * 2) * NCH + (size_t)blk * BPO) * T_OUT + (size_t)t;
#pragma unroll
  for (int r = 0; r < 8; ++r) {
    int row = mbase + r;
    if (row < NFILT) {
      size_t o = base0 + (size_t)row * T_OUT +
                 (row >= BPO ? (size_t)(NCH - BPO) * T_OUT : (size_t)0);
      out[o] = c[r];
    }
  }
}

// ---------------------------------------------------------------------------
extern "C" void kernel_launch(void* const* d_in, const int* in_sizes, int n_in,
                              void* d_out, int out_size, void* d_ws, size_t ws_size,
                              hipStream_t stream) {
  (void)in_sizes; (void)n_in; (void)out_size; (void)ws_size;
  const float* x    = (const float*)d_in[0];   // (16, 524288)
  const float* kern = (const float*)d_in[1];   // (72, 1, 271)
  const float* num  = (const float*)d_in[2];   // (11,)
  const float* den  = (const float*)d_in[3];   // (11,)
  float* out = (float*)d_out;                  // (16, 2, 288, 2048)
  float* ws  = (float*)d_ws;

  float* G       = ws;          // 512 floats reserved for 129 FIR taps
  float* sigbase = ws + 512;    // downsampled signals, levels 1..7

  // Stage A: filtfilt-equivalent FIR taps.
  fir_taps_kernel<<<1, 64, 0, stream>>>(num, den, G);

  // Stage B: decimation chain (level l has length L0 >> l).
  const float* prev = x;
  float* levptr[OCTAVES];
  levptr[0] = nullptr;
  float* p  = sigbase;
  int Lin   = L0;
  for (int l = 1; l < OCTAVES; ++l) {
    int Lout = Lin >> 1;
    dim3 grid((Lout + 255) / 256, BATCH);
    downsample_kernel<<<grid, 256, 0, stream>>>(prev, p, G, Lin, Lout);
    levptr[l] = p;
    prev = p;
    p   += (size_t)BATCH * Lout;
    Lin  = Lout;
  }

  // Stage C: WMMA CQT per level. Level l -> output octave block 7-l.
  int L = L0, hop = 256;
  for (int l = 0; l < OCTAVES; ++l) {
    const float* sig = (l == 0) ? x : levptr[l];
    dim3 grid(T_OUT / 32, 5 /* ceil(72/16) */, BATCH);
    cqt_wmma_kernel<<<grid, 64, 0, stream>>>(sig, kern, out, L, hop, 7 - l);
    L >>= 1; hop >>= 1;
  }
}